// MultiScaleRetention_60413009985546
// MI455X (gfx1250) — compile-verified
//
#include <hip/hip_runtime.h>
#include <hip/hip_bf16.h>
#include <math.h>

// MultiScaleRetention for MI455X (gfx1250), fp32 via V_WMMA_F32_16X16X4_F32,
// staged through LDS with CDNA5 GLOBAL_LOAD_ASYNC_TO_LDS_B128 (ASYNCcnt).
//
// Decay folding:  scores*D = (Q*g^s)(K*g^-t)^T ; linear attention assoc:
//   ret = Q' @ (K'^T @ V)   -> two 128-wide GEMMs per (b,h) instead of SxS.
//
// Pipeline (all on `stream`):
//  1) k_qkv     : Q'=(x_h Wq)*g^s, K'=(x_h Wk)*g^-s, V=x_h Wv     [B,H,S,128]
//  2) k_state   : M = K'^T V  (split-K over up to 8 chunks)       [B,H,128,128]
//  3) k_state_reduce : sum split-K partials
//  4) k_ret     : y_h = Q' M  -> concat heads into y [B,S,1024]
//  5) k_gnstats : per (b,group) mean / rsqrt(var+eps) (groups=128, 8 ch each)
//  6) k_gate    : G = ReLU(x Wg)                                  [B,S,1024]
//  7) k_outprep : G *= groupnorm(y)  (elementwise, bandwidth-bound)
//  8) k_out     : out = G_normed @ Wo
//
// GEMM core: 128x128 block tile, 8 waves (4Mx2N), wave tile 32x64.
// Double-buffered LDS; next k-stage tiles stream in via async-to-LDS while the
// current stage feeds the WMMA pipe from DS.

typedef __attribute__((ext_vector_type(2))) float v2f;
typedef __attribute__((ext_vector_type(8))) float v8f;

#define BB 8
#define SS 1024
#define DD 1024
#define HD 128
#define NH 8

// 16-byte global -> LDS async copy (CDNA5 TDM-adjacent async path, ASYNCcnt).
__device__ __forceinline__ void async_copy16(unsigned lds_off, const float* g)
{
  asm volatile("global_load_async_to_lds_b128 %0, %1, off"
               :: "v"(lds_off), "v"((unsigned long long)(uintptr_t)g)
               : "memory");
}
__device__ __forceinline__ void wait_async0()
{
  asm volatile("s_wait_asynccnt 0x0" ::: "memory");
}

// ---------------------------------------------------------------------------
// AMAP==0: A element (m,k) at A[m*lda + k]   (row-major; rows async-copied)
// AMAP==1: A element (m,k) at A[k*lda + m]   (k-major rows async-copied; used
//          for K'^T so the transpose is resolved at the DS operand fetch)
// ---------------------------------------------------------------------------
template<int AMAP>
__device__ __forceinline__ void gemm_core(const float* __restrict__ A, int lda,
                                          const float* __restrict__ Bm, int ldb,
                                          int K, int m0, int n0, v8f (&acc)[2][4])
{
  constexpr int APITCH = (AMAP == 0) ? 20 : 132;   // *4B: multiple of 16B
  constexpr int AROWS  = (AMAP == 0) ? 128 : 16;
  __shared__ __align__(16) float As[2][AROWS * APITCH];
  __shared__ __align__(16) float Bs[2][16 * 132];

  const int tid  = threadIdx.x;
  const int wv   = tid >> 5, lane = tid & 31;
  const int mw   = (wv & 3) * 32, nw = (wv >> 2) * 64;
  const int idx  = lane & 15, hi = lane >> 4;      // ISA 7.12.2 operand layout

  const v8f zero = {0.f,0.f,0.f,0.f,0.f,0.f,0.f,0.f};
  #pragma unroll
  for (int i = 0; i < 2; ++i)
    #pragma unroll
    for (int j = 0; j < 4; ++j) acc[i][j] = zero;

  const unsigned asBase = (unsigned)(uintptr_t)&As[0][0];
  const unsigned bsBase = (unsigned)(uintptr_t)&Bs[0][0];
  constexpr unsigned asStride = AROWS * APITCH * 4u;
  constexpr unsigned bsStride = 16 * 132 * 4u;

  auto issueStage = [&](int kk, int buf) {
    if (AMAP == 0) {                       // 128 rows x 64B = 512 chunks
      #pragma unroll
      for (int i = 0; i < 2; ++i) {
        const int q = tid + 256 * i;
        const int m = q >> 2, cp = (q & 3) * 4;
        async_copy16(asBase + buf * asStride + (unsigned)(m * APITCH + cp) * 4u,
                     A + (size_t)(m0 + m) * lda + kk + cp);
      }
    } else {                               // 16 k-rows x 512B = 512 chunks
      #pragma unroll
      for (int i = 0; i < 2; ++i) {
        const int q = tid + 256 * i;
        const int k = q >> 5, cp = (q & 31) * 4;
        async_copy16(asBase + buf * asStride + (unsigned)(k * APITCH + cp) * 4u,
                     A + (size_t)(kk + k) * lda + m0 + cp);
      }
    }
    #pragma unroll
    for (int i = 0; i < 2; ++i) {          // B: 16 k-rows x 512B = 512 chunks
      const int q = tid + 256 * i;
      const int k = q >> 5, cp = (q & 31) * 4;
      async_copy16(bsBase + buf * bsStride + (unsigned)(k * 132 + cp) * 4u,
                   Bm + (size_t)(kk + k) * ldb + n0 + cp);
    }
  };

  const int nstages = K >> 4;
  issueStage(0, 0);
  wait_async0();
  __syncthreads();

  for (int s = 0; s < nstages; ++s) {
    const int buf = s & 1;
    if (s + 1 < nstages) issueStage((s + 1) << 4, 1 - buf);
    if (s + 2 < nstages)  // keep a prefetch two stages out (global_prefetch_b8)
      __builtin_prefetch(&Bm[(size_t)(((s + 2) << 4) + (tid >> 7)) * ldb
                             + n0 + (tid & 127)], 0, 3);

    #pragma unroll
    for (int kw = 0; kw < 16; kw += 4) {
      const int kc = kw + 2 * hi;          // even
      v2f a[2], bf[4];
      #pragma unroll
      for (int i = 0; i < 2; ++i) {
        const int m = mw + 16 * i + idx;
        if (AMAP == 0) {
          a[i] = *(const v2f*)&As[buf][m * APITCH + kc];
        } else {
          a[i].x = As[buf][kc * APITCH + m];
          a[i].y = As[buf][(kc + 1) * APITCH + m];
        }
      }
      #pragma unroll
      for (int j = 0; j < 4; ++j) {
        const int n = nw + 16 * j + idx;
        bf[j].x = Bs[buf][kc * 132 + n];
        bf[j].y = Bs[buf][(kc + 1) * 132 + n];
      }
      #pragma unroll
      for (int i = 0; i < 2; ++i)
        #pragma unroll
        for (int j = 0; j < 4; ++j)
          acc[i][j] = __builtin_amdgcn_wmma_f32_16x16x4_f32(
              false, a[i], false, bf[j], (short)0, acc[i][j], false, false);
    }

    wait_async0();        // this wave's next-stage copies have landed in LDS
    __syncthreads();      // ...and every other wave's too
  }
}

// C/D layout (ISA 7.12.2): VGPR r -> row r + 8*hi within a 16x16 tile.
template<typename ST>
__device__ __forceinline__ void store_acc(v8f (&acc)[2][4], ST st)
{
  const int tid = threadIdx.x;
  const int wv  = tid >> 5, lane = tid & 31;
  const int mw  = (wv & 3) * 32, nw = (wv >> 2) * 64;
  const int idx = lane & 15, hi = lane >> 4;
  #pragma unroll
  for (int i = 0; i < 2; ++i)
    #pragma unroll
    for (int j = 0; j < 4; ++j)
      #pragma unroll
      for (int r = 0; r < 8; ++r)
        st(mw + 16*i + r + 8*hi, nw + 16*j + idx, acc[i][j][r]);
}

// ---------------------------------------------------------------------------
__global__ __launch_bounds__(256) void k_qkv(const float* __restrict__ x,
    const float* __restrict__ Wqg, const float* __restrict__ Wkg,
    const float* __restrict__ Wvg,
    float* __restrict__ Qp, float* __restrict__ Kp, float* __restrict__ Vp)
{
  const int s0 = blockIdx.x * 128;
  const int bh = blockIdx.y;
  const int h = bh & 7;
  const int b = bh >> 3;
  const int which = blockIdx.z;     // 0=Q 1=K 2=V

  const float* A = x + (size_t)b * SS * DD + (size_t)h * HD;
  const float* W = (which == 0 ? Wqg : which == 1 ? Wkg : Wvg) + (size_t)h * HD * HD;
  float* outp    = (which == 0 ? Qp  : which == 1 ? Kp  : Vp)  + (size_t)bh * SS * HD;

  v8f acc[2][4];
  gemm_core<0>(A, DD, W, HD, HD, s0, 0, acc);

  const float gam = 1.f - exp2f(-5.f - (float)h);
  const float l2g = log2f(gam);
  const float sgn = (which == 0) ? 1.f : -1.f;   // Q: g^s ; K: g^-s
  store_acc(acc, [&](int ml, int n, float v) {
    const float f = (which == 2) ? 1.f : exp2f(sgn * l2g * (float)(s0 + ml));
    outp[(size_t)(s0 + ml) * HD + n] = v * f;
  });
}

__global__ __launch_bounds__(256) void k_state(const float* __restrict__ Kp,
    const float* __restrict__ Vp, float* __restrict__ Mpart, int ck)
{
  const int bh = blockIdx.x;
  const int chunk = blockIdx.y;
  const int kbase = chunk * ck;
  const float* Ka = Kp + (size_t)bh * SS * HD + (size_t)kbase * HD;  // [k][m]
  const float* Va = Vp + (size_t)bh * SS * HD + (size_t)kbase * HD;
  v8f acc[2][4];
  gemm_core<1>(Ka, HD, Va, HD, ck, 0, 0, acc);
  float* outp = Mpart + ((size_t)chunk * (BB * NH) + bh) * HD * HD;
  store_acc(acc, [&](int ml, int n, float v) { outp[(size_t)ml * HD + n] = v; });
}

__global__ __launch_bounds__(256) void k_state_reduce(const float* __restrict__ Mpart,
    float* __restrict__ Mst, int nchunks)
{
  const size_t i = (size_t)blockIdx.x * 256 + threadIdx.x;
  float s = 0.f;
  for (int cidx = 0; cidx < nchunks; ++cidx)
    s += Mpart[(size_t)cidx * (BB * NH * HD * HD) + i];
  Mst[i] = s;
}

__global__ __launch_bounds__(256) void k_ret(const float* __restrict__ Qp,
    const float* __restrict__ Mst, float* __restrict__ Yb)
{
  const int s0 = blockIdx.x * 128;
  const int bh = blockIdx.y;
  const int b = bh >> 3, h = bh & 7;
  const float* A  = Qp  + (size_t)bh * SS * HD;
  const float* Bm = Mst + (size_t)bh * HD * HD;
  v8f acc[2][4];
  gemm_core<0>(A, HD, Bm, HD, HD, s0, 0, acc);
  store_acc(acc, [&](int ml, int n, float v) {
    Yb[((size_t)b * SS + (s0 + ml)) * DD + (size_t)h * HD + n] = v;
  });
}

__global__ __launch_bounds__(256) void k_gnstats(const float* __restrict__ Yb,
                                                 float* __restrict__ Stats)
{
  // blockIdx.x = b*128 + g ; group g = channels [8g, 8g+8), reduce over S and 8.
  const int b = blockIdx.x >> 7, g = blockIdx.x & 127;
  const int tid = threadIdx.x;
  const int ch = (g << 3) + (tid & 7);
  float s1 = 0.f, s2 = 0.f;
  for (int s = tid >> 3; s < SS; s += 32) {
    const float v = Yb[((size_t)b * SS + s) * DD + ch];
    s1 += v; s2 += v * v;
  }
  __shared__ float r1[256], r2[256];
  r1[tid] = s1; r2[tid] = s2;
  __syncthreads();
  for (int off = 128; off > 0; off >>= 1) {
    if (tid < off) { r1[tid] += r1[tid + off]; r2[tid] += r2[tid + off]; }
    __syncthreads();
  }
  if (tid == 0) {
    const float mean = r1[0] * (1.f / 8192.f);
    const float var  = r2[0] * (1.f / 8192.f) - mean * mean;
    Stats[blockIdx.x]        = mean;
    Stats[1024 + blockIdx.x] = rsqrtf(var + 1e-3f);
  }
}

__global__ __launch_bounds__(256) void k_gate(const float* __restrict__ x,
    const float* __restrict__ Wg, float* __restrict__ Gb)
{
  const int n0 = blockIdx.x * 128;
  const int m0 = blockIdx.y * 128;
  v8f acc[2][4];
  gemm_core<0>(x, DD, Wg, DD, DD, m0, n0, acc);
  store_acc(acc, [&](int ml, int n, float v) {
    Gb[(size_t)(m0 + ml) * DD + (n0 + n)] = fmaxf(v, 0.f);
  });
}

// G[m,k] *= groupnorm(y)[m,k]  (in-place; G freshly produced by k_gate)
__global__ __launch_bounds__(256) void k_outprep(float* __restrict__ Gb,
    const float* __restrict__ Yb, const float* __restrict__ Stats,
    const float* __restrict__ gn_g, const float* __restrict__ gn_b)
{
  size_t i = (size_t)blockIdx.x * 1024 + threadIdx.x;
  #pragma unroll
  for (int t = 0; t < 4; ++t, i += 256) {
    const int k = (int)(i & 1023);
    const int m = (int)(i >> 10);
    const int b = m >> 10, g = k >> 3;
    const float mean = Stats[b * 128 + g];
    const float rv   = Stats[1024 + b * 128 + g];
    const float yn = (Yb[i] - mean) * rv * gn_g[k] + gn_b[k];
    Gb[i] *= yn;
  }
}

__global__ __launch_bounds__(256) void k_out(const float* __restrict__ Gb,
    const float* __restrict__ Wo, float* __restrict__ outp)
{
  const int n0 = blockIdx.x * 128;
  const int m0 = blockIdx.y * 128;
  v8f acc[2][4];
  gemm_core<0>(Gb, DD, Wo, DD, DD, m0, n0, acc);
  store_acc(acc, [&](int ml, int n, float v) {
    outp[(size_t)(m0 + ml) * DD + (n0 + n)] = v;
  });
}

// ---------------------------------------------------------------------------
extern "C" void kernel_launch(void* const* d_in, const int* in_sizes, int n_in,
                              void* d_out, int out_size, void* d_ws, size_t ws_size,
                              hipStream_t stream)
{
  (void)in_sizes; (void)n_in; (void)out_size;

  const float* x    = (const float*)d_in[0];
  const float* Wq   = (const float*)d_in[1];
  const float* Wk   = (const float*)d_in[2];
  const float* Wv   = (const float*)d_in[3];
  const float* Wg   = (const float*)d_in[4];
  const float* Wo   = (const float*)d_in[5];
  const float* gn_g = (const float*)d_in[6];
  const float* gn_b = (const float*)d_in[7];
  float* outp = (float*)d_out;

  // Workspace carve-up (floats). Base ~43.0M floats (~164 MB) + split-K partials.
  float* ws  = (float*)d_ws;
  const size_t NQ   = (size_t)BB * NH * SS * HD;   // 8,388,608
  const size_t NM   = (size_t)BB * NH * HD * HD;   // 1,048,576
  float* Qp    = ws;
  float* Kp    = Qp  + NQ;
  float* Vp    = Kp  + NQ;
  float* Mst   = Vp  + NQ;
  float* Yb    = Mst + NM;
  float* Gb    = Yb  + (size_t)BB * SS * DD;
  float* Stats = Gb  + (size_t)BB * SS * DD;       // 2048
  float* Mpart = Stats + 2048;

  // Split-K factor for k_state chosen from available workspace (4 MB/chunk).
  const size_t baseFloats = (size_t)(Mpart - ws);
  const size_t avail = (ws_size / sizeof(float) > baseFloats)
                           ? ws_size / sizeof(float) - baseFloats : 0;
  int nchunks = 1;
  if (avail >= 8 * NM) nchunks = 8;
  else if (avail >= 4 * NM) nchunks = 4;
  else if (avail >= 2 * NM) nchunks = 2;
  float* MpartUse = (avail >= NM) ? Mpart : Mst;   // degenerate fallback
  const int ck = SS / nchunks;                     // multiple of 16

  const dim3 blk(256);
  k_qkv         <<<dim3(SS / 128, BB * NH, 3), blk, 0, stream>>>(x, Wq, Wk, Wv, Qp, Kp, Vp);
  k_state       <<<dim3(BB * NH, nchunks),     blk, 0, stream>>>(Kp, Vp, MpartUse, ck);
  k_state_reduce<<<dim3((unsigned)(NM / 256)), blk, 0, stream>>>(MpartUse, Mst, nchunks);
  k_ret         <<<dim3(SS / 128, BB * NH),    blk, 0, stream>>>(Qp, Mst, Yb);
  k_gnstats     <<<dim3(BB * 128),             blk, 0, stream>>>(Yb, Stats);
  k_gate        <<<dim3(DD / 128, BB * SS / 128), blk, 0, stream>>>(x, Wg, Gb);
  k_outprep     <<<dim3((unsigned)((size_t)BB * SS * DD / 1024)), blk, 0, stream>>>(
                    Gb, Yb, Stats, gn_g, gn_b);
  k_out         <<<dim3(DD / 128, BB * SS / 128), blk, 0, stream>>>(Gb, Wo, outp);
}